// TimedGCN_7224134992216
// MI455X (gfx1250) — compile-verified
//
#include <hip/hip_runtime.h>
#include <hip/hip_bf16.h>

typedef float v2f __attribute__((ext_vector_type(2)));
typedef float v8f __attribute__((ext_vector_type(8)));

#define N_NODES 50000
#define N_EDGES 800000
#define F_IN    128
#define F_HID   256
#define F_OUT   40
#define M_TILES (N_NODES / 16)   // 3125, exact

__device__ __forceinline__ void atomic_add_f32(float* p, float v) {
    __hip_atomic_fetch_add(p, v, __ATOMIC_RELAXED, __HIP_MEMORY_SCOPE_AGENT);
}

// ---------------------------------------------------------------------------
// GEMM1: H1 = X @ W1   (X:[50000,128], W1:[128,256] row-major, H1:[50000,256])
// Block = 256 threads = 8 waves. Each wave: 16 rows x 64 cols, K swept by 4.
// W1 64-col panel in LDS with K-pair interleaving:
//   sW[(k>>1)*128 + c*2 + (k&1)]  ->  (W[k][c], W[k+1][c]) is one aligned b64.
// ---------------------------------------------------------------------------
__global__ void __launch_bounds__(256)
gcn_gemm1(const float* __restrict__ X, const float* __restrict__ W,
          float* __restrict__ H) {
    __shared__ float sW[F_IN * 64];
    const int t     = threadIdx.x;
    const int cBase = blockIdx.y * 64;
#pragma unroll
    for (int i = 0; i < (F_IN * 64) / 256; ++i) {
        int e = i * 256 + t;
        int r = e >> 6;                       // 0..127
        int c = e & 63;                       // 0..63
        sW[(r >> 1) * 128 + c * 2 + (r & 1)] = W[r * F_HID + cBase + c];
    }
    __syncthreads();

    const int wave  = t >> 5;
    const int lane  = t & 31;
    const int laneM = lane & 15;
    const int half  = lane >> 4;              // 0: K=0,1  1: K=2,3
    const int kOff  = half << 1;
    const int mTile = blockIdx.x * 8 + wave;
    if (mTile >= M_TILES) return;

    const float* aPtr = X + (size_t)(mTile * 16 + laneM) * F_IN + kOff;
    // per-lane LDS base: pair row (k>>1) = kk/2 + half, col entry c*2
    const float* bBase = sW + (size_t)half * 128 + laneM * 2;
    v8f acc0 = {}, acc1 = {}, acc2 = {}, acc3 = {};

    for (int kk = 0; kk < F_IN; kk += 4) {
        v2f a = *(const v2f*)(aPtr + kk);                 // A[row][kk+kOff..+1]
        const float* bp = bBase + (kk >> 1) * 128;
        v2f b0 = *(const v2f*)(bp +  0);                  // single ds_load_b64
        v2f b1 = *(const v2f*)(bp + 32);
        v2f b2 = *(const v2f*)(bp + 64);
        v2f b3 = *(const v2f*)(bp + 96);
        acc0 = __builtin_amdgcn_wmma_f32_16x16x4_f32(false, a, false, b0, (short)0, acc0, false, false);
        acc1 = __builtin_amdgcn_wmma_f32_16x16x4_f32(false, a, false, b1, (short)0, acc1, false, false);
        acc2 = __builtin_amdgcn_wmma_f32_16x16x4_f32(false, a, false, b2, (short)0, acc2, false, false);
        acc3 = __builtin_amdgcn_wmma_f32_16x16x4_f32(false, a, false, b3, (short)0, acc3, false, false);
    }

    // C/D layout: VGPR i -> M=i (lanes 0-15), M=i+8 (lanes 16-31), N=laneM
    const int rowBase = mTile * 16 + half * 8;
    float* hBase = H + (size_t)rowBase * F_HID + cBase + laneM;
#pragma unroll
    for (int i = 0; i < 8; ++i) {
        float* p = hBase + (size_t)i * F_HID;
        p[0]  = acc0[i];
        p[16] = acc1[i];
        p[32] = acc2[i];
        p[48] = acc3[i];
    }
}

// ---------------------------------------------------------------------------
// GEMM2: H3 = H2 @ W2  (H2:[50000,256], W2:[256,40] padded to 48 cols)
// Same K-pair-interleaved LDS layout: sW[(k>>1)*96 + c*2 + (k&1)].
// ---------------------------------------------------------------------------
__global__ void __launch_bounds__(256)
gcn_gemm2(const float* __restrict__ H2, const float* __restrict__ W,
          float* __restrict__ O) {
    __shared__ float sW[F_HID * 48];               // 48 KB
    const int t = threadIdx.x;
#pragma unroll
    for (int i = 0; i < (F_HID * 48) / 256; ++i) {
        int e = i * 256 + t;
        int r = e / 48;
        int c = e % 48;
        sW[(r >> 1) * 96 + c * 2 + (r & 1)] = (c < F_OUT) ? W[r * F_OUT + c] : 0.0f;
    }
    __syncthreads();

    const int wave  = t >> 5;
    const int lane  = t & 31;
    const int laneM = lane & 15;
    const int half  = lane >> 4;
    const int kOff  = half << 1;
    const int mTile = blockIdx.x * 8 + wave;
    if (mTile >= M_TILES) return;

    const float* aPtr  = H2 + (size_t)(mTile * 16 + laneM) * F_HID + kOff;
    const float* bBase = sW + (size_t)half * 96 + laneM * 2;
    v8f acc0 = {}, acc1 = {}, acc2 = {};

    for (int kk = 0; kk < F_HID; kk += 4) {
        v2f a = *(const v2f*)(aPtr + kk);
        const float* bp = bBase + (kk >> 1) * 96;
        v2f b0 = *(const v2f*)(bp +  0);
        v2f b1 = *(const v2f*)(bp + 32);
        v2f b2 = *(const v2f*)(bp + 64);
        acc0 = __builtin_amdgcn_wmma_f32_16x16x4_f32(false, a, false, b0, (short)0, acc0, false, false);
        acc1 = __builtin_amdgcn_wmma_f32_16x16x4_f32(false, a, false, b1, (short)0, acc1, false, false);
        acc2 = __builtin_amdgcn_wmma_f32_16x16x4_f32(false, a, false, b2, (short)0, acc2, false, false);
    }

    const int rowBase = mTile * 16 + half * 8;
#pragma unroll
    for (int i = 0; i < 8; ++i) {
        float* p = O + (size_t)(rowBase + i) * F_OUT + laneM;
        p[0]  = acc0[i];
        p[16] = acc1[i];
        if (laneM < 8) p[32] = acc2[i];            // cols 32..39 only
    }
}

// ---------------------------------------------------------------------------
// Scatter layer 1: AGG[dst] += H1[src], 256 floats/edge, 64 threads/edge.
// ---------------------------------------------------------------------------
__global__ void __launch_bounds__(256)
gcn_scatter1(const float* __restrict__ H1, const int* __restrict__ srcIdx,
             const int* __restrict__ dstIdx, float* __restrict__ AGG) {
    const int tid = blockIdx.x * 256 + threadIdx.x;
    const int e   = tid >> 6;
    const int c4  = (tid & 63) << 2;
    const int s = srcIdx[e];
    const int d = dstIdx[e];
    const float4 v = *(const float4*)(H1 + (size_t)s * F_HID + c4);
    float* p = AGG + (size_t)d * F_HID + c4;
    atomic_add_f32(p + 0, v.x);
    atomic_add_f32(p + 1, v.y);
    atomic_add_f32(p + 2, v.z);
    atomic_add_f32(p + 3, v.w);
}

// H2 = relu(AGG + b1), in place.
__global__ void __launch_bounds__(256)
gcn_bias_relu(float* __restrict__ AGG, const float* __restrict__ b) {
    const int tid = blockIdx.x * 256 + threadIdx.x;     // exactly N*F_HID
    const float v = AGG[tid] + b[tid & (F_HID - 1)];
    AGG[tid] = v > 0.0f ? v : 0.0f;
}

// out = broadcast(b2)
__global__ void __launch_bounds__(256)
gcn_init_out(float* __restrict__ O, const float* __restrict__ b) {
    const int tid = blockIdx.x * 256 + threadIdx.x;
    if (tid < N_NODES * F_OUT) O[tid] = b[tid % F_OUT];
}

// Scatter layer 2: out[dst] += H3[src], 40 floats/edge, 10 float4 slots/edge.
__global__ void __launch_bounds__(256)
gcn_scatter2(const float* __restrict__ H3, const int* __restrict__ srcIdx,
             const int* __restrict__ dstIdx, float* __restrict__ O) {
    const int tid = blockIdx.x * 256 + threadIdx.x;     // exactly E*10
    const int e   = tid / 10;
    const int c4  = (tid % 10) << 2;
    const int s = srcIdx[e];
    const int d = dstIdx[e];
    const float4 v = *(const float4*)(H3 + (size_t)s * F_OUT + c4);
    float* p = O + (size_t)d * F_OUT + c4;
    atomic_add_f32(p + 0, v.x);
    atomic_add_f32(p + 1, v.y);
    atomic_add_f32(p + 2, v.z);
    atomic_add_f32(p + 3, v.w);
}

extern "C" void kernel_launch(void* const* d_in, const int* in_sizes, int n_in,
                              void* d_out, int out_size, void* d_ws, size_t ws_size,
                              hipStream_t stream) {
    const float* x  = (const float*)d_in[0];
    const int*   ei = (const int*)d_in[1];
    const float* W1 = (const float*)d_in[2];
    const float* b1 = (const float*)d_in[3];
    const float* W2 = (const float*)d_in[4];
    const float* b2 = (const float*)d_in[5];
    float* out = (float*)d_out;

    const int* srcIdx = ei;
    const int* dstIdx = ei + N_EDGES;

    // workspace layout: H1 (12.8M) | AGG/H2 (12.8M) | H3 (2M) floats
    float* H1  = (float*)d_ws;
    float* AGG = H1  + (size_t)N_NODES * F_HID;
    float* H3  = AGG + (size_t)N_NODES * F_HID;

    hipMemsetAsync(AGG, 0, sizeof(float) * (size_t)N_NODES * F_HID, stream);

    dim3 g1((M_TILES + 7) / 8, F_HID / 64);
    gcn_gemm1<<<g1, 256, 0, stream>>>(x, W1, H1);

    gcn_scatter1<<<(N_EDGES * 64) / 256, 256, 0, stream>>>(H1, srcIdx, dstIdx, AGG);
    gcn_bias_relu<<<(N_NODES * F_HID) / 256, 256, 0, stream>>>(AGG, b1);

    gcn_gemm2<<<(M_TILES + 7) / 8, 256, 0, stream>>>(AGG, W2, H3);

    gcn_init_out<<<(N_NODES * F_OUT + 255) / 256, 256, 0, stream>>>(out, b2);
    gcn_scatter2<<<(N_EDGES * 10) / 256, 256, 0, stream>>>(H3, srcIdx, dstIdx, out);
}